// BandPass_46437186404846
// MI455X (gfx1250) — compile-verified
//
#include <hip/hip_runtime.h>
#include <cmath>
#include <complex>
#include <algorithm>

// ---------------- problem constants (match reference) ----------------
#define ROWS 768            // 64*12 independent signals
#define TLEN 30000          // samples per signal
#define PADL 33             // 3*(2*ORDER+1)
#define TEXT (TLEN + 2*PADL)   // 30066 extended length
#define NBLK 1880              // ceil(TEXT/16)
#define TPAD (NBLK*16)         // 30080 padded extended length
#define NGRP (ROWS/16)         // 48 groups of 16 rows (one wave each)

typedef __attribute__((ext_vector_type(2))) float v2f;
typedef __attribute__((ext_vector_type(8))) float v8f;

// Per-section data (derived on host in double, exact):
//   h[t]  : impulse response (h[0]=d=b0, h[t]=C A^{t-1} B)  -> Toeplitz A-matrix
//   g1/g2 : first row of A^t (state->output response), folded into WMMA as 5th chunk
//   b1,b2,a1,a2 : raw biquad coefs (block-end state from last 2 samples)
//   zi1/2 : lfilter_zi initial-state coefficients (scaled by x0 in-kernel)
struct Sec {
  float h[16];
  float g1[16], g2[16];
  float b1, b2, a1, a2;
  float zi1, zi2;
};
struct Coef { Sec s[5]; };

// ---------------- device ----------------

__device__ __forceinline__ float load_edge0(const float* __restrict__ xin, int t) {
  // filtfilt odd edge extension, built on the fly (pass 0 only)
  if (t >= TEXT) return 0.f;                    // block tail padding (causal -> harmless)
  if (t < PADL) return 2.f * xin[0] - xin[PADL - t];
  int i = t - PADL;
  if (i < TLEN) return xin[i];
  return 2.f * xin[TLEN - 1] - xin[2*TLEN + PADL - 2 - t];
}

__global__ __launch_bounds__(32)
void BandPass_filtfilt_wmma(const float* __restrict__ in,
                            float* __restrict__ out,
                            float* __restrict__ ws,
                            Coef cf) {
  const int g    = blockIdx.x;        // row group (16 rows)
  const int lane = threadIdx.x & 31;
  const int r    = lane & 15;         // row-in-group == N column == A-matrix M row
  const int hi   = lane >> 4;         // lane half (K/M split per WMMA layouts)
  const int row  = g * 16 + r;
  const float* xin = in + (size_t)row * TLEN;
  float* wsg = ws + (size_t)g * ((size_t)TPAD * 16);

  // ---- per-lane constant precompute (once): 5*(4+1) v2f = 50 VGPRs total ----
  // A-matrix (16x4 f32): lanes hold M=r; VGPR0 = K=(0|2)+4kk, VGPR1 = K=(1|3)+4kk.
  v2f hA[5][4];   // lower-triangular Toeplitz chunks of h
  v2f sA[5];      // state-response chunk: columns [g1 g2 0 0]
  #pragma unroll
  for (int k = 0; k < 5; ++k) {
    #pragma unroll
    for (int kk = 0; kk < 4; ++kk) {
      int j0 = 4*kk + 2*hi, j1 = j0 + 1;
      hA[k][kk].x = (j0 <= r) ? cf.s[k].h[r - j0] : 0.f;
      hA[k][kk].y = (j1 <= r) ? cf.s[k].h[r - j1] : 0.f;
    }
    sA[k].x = hi ? 0.f : cf.s[k].g1[r];   // K=0 col = g1, K=2 col = 0
    sA[k].y = hi ? 0.f : cf.s[k].g2[r];   // K=1 col = g2, K=3 col = 0
  }

  for (int pass = 0; pass < 2; ++pass) {
    // lfilter_zi scaling: x0 = first sample of this pass's cascade input
    float x0 = (pass == 0) ? (2.f * xin[0] - xin[PADL])
                           : wsg[(size_t)(TEXT - 1) * 16 + r];
    float z1[5], z2[5];
    #pragma unroll
    for (int k = 0; k < 5; ++k) { z1[k] = cf.s[k].zi1 * x0; z2[k] = cf.s[k].zi2 * x0; }

    for (int blk = 0; blk < NBLK; ++blk) {
      const int tb = blk * 16;

      // ---- load B-matrix (16x16 over four 4x16 K-chunks) ----
      // lane n=r; VGPR0 = K(0|2)+4kk, VGPR1 = K(1|3)+4kk
      v2f bm[4];
      if (pass == 0) {
        if (tb >= PADL && tb + 16 <= PADL + TLEN) {        // interior: branch-free
          const float* bp = xin + (tb - PADL);
          int pf = tb + 2048;
          if (pf < TLEN) __builtin_prefetch(xin + pf, 0, 3);   // global_prefetch_b8
          #pragma unroll
          for (int kk = 0; kk < 4; ++kk) {
            int o = 4*kk + 2*hi;
            bm[kk].x = bp[o];
            bm[kk].y = bp[o + 1];
          }
        } else {                                           // 3 head + 3 tail blocks
          #pragma unroll
          for (int kk = 0; kk < 4; ++kk) {
            int t0 = tb + 4*kk + 2*hi;
            bm[kk].x = load_edge0(xin, t0);
            bm[kk].y = load_edge0(xin, t0 + 1);
          }
        }
      } else {
        if (tb + 16 <= TEXT) {                             // interior reversed read
          const float* bp = wsg + (size_t)(TEXT - 1 - tb) * 16 + r;
          #pragma unroll
          for (int kk = 0; kk < 4; ++kk) {
            int o = 4*kk + 2*hi;
            bm[kk].x = bp[-(o    ) * 16];
            bm[kk].y = bp[-(o + 1) * 16];
          }
        } else {                                           // last block only
          #pragma unroll
          for (int kk = 0; kk < 4; ++kk) {
            int t0 = tb + 4*kk + 2*hi;
            bm[kk].x = (t0     < TEXT) ? wsg[(size_t)(TEXT - 1 - t0) * 16 + r] : 0.f;
            bm[kk].y = (t0 + 1 < TEXT) ? wsg[(size_t)(TEXT - 2 - t0) * 16 + r] : 0.f;
          }
        }
      }

      v8f c;
      #pragma unroll
      for (int k = 0; k < 5; ++k) {
        // ---- Y = G*s0 + H*X : five chained f32 WMMAs (state folded in) ----
        v2f sb;
        sb.x = hi ? 0.f : z1[k];      // B rows: K0=z1, K1=z2, K2=K3=0
        sb.y = hi ? 0.f : z2[k];
        #pragma unroll
        for (int v = 0; v < 8; ++v) c[v] = 0.f;
        c = __builtin_amdgcn_wmma_f32_16x16x4_f32(false, sA[k], false, sb,
                                                  (short)0, c, false, false);
        #pragma unroll
        for (int kk = 0; kk < 4; ++kk)
          c = __builtin_amdgcn_wmma_f32_16x16x4_f32(false, hA[k][kk], false, bm[kk],
                                                    (short)0, c, false, false);

        // ---- block-end state, exact TDF-II from last two samples ----
        // hi=1 half holds x14,x15 (bm[3]) and y14,y15 (c[6],c[7])
        {
          float xA = bm[3].x, xB = bm[3].y;           // x14, x15 on hi=1
          float yA = c[6],    yB = c[7];              // y14, y15 on hi=1
          float nz2 = cf.s[k].b2 * xB - cf.s[k].a2 * yB;
          float nz1 = cf.s[k].b1 * xB - cf.s[k].a1 * yB
                    + cf.s[k].b2 * xA - cf.s[k].a2 * yA;
          float sz1 = __shfl_xor(nz1, 16, 32);        // broadcast hi=1 -> hi=0
          float sz2 = __shfl_xor(nz2, 16, 32);
          z1[k] = hi ? nz1 : sz1;
          z2[k] = hi ? nz2 : sz2;
        }

        // ---- D-layout -> B-layout for next section (cross half-lane exchange) ----
        if (k < 4) {
          float s0 = hi ? c[0] : c[2];
          float s1 = hi ? c[1] : c[3];
          float s2 = hi ? c[4] : c[6];
          float s3 = hi ? c[5] : c[7];
          float q0 = __shfl_xor(s0, 16, 32);
          float q1 = __shfl_xor(s1, 16, 32);
          float q2 = __shfl_xor(s2, 16, 32);
          float q3 = __shfl_xor(s3, 16, 32);
          float oA0 = hi ? c[2] : c[0];
          float oA1 = hi ? c[3] : c[1];
          float oB0 = hi ? c[6] : c[4];
          float oB1 = hi ? c[7] : c[5];
          bm[0].x = hi ? q0 : oA0;  bm[0].y = hi ? q1 : oA1;   // times 4kk+2hi(+1)
          bm[1].x = hi ? q2 : oB0;  bm[1].y = hi ? q3 : oB1;
          bm[2].x = hi ? oA0 : q0;  bm[2].y = hi ? oA1 : q1;
          bm[3].x = hi ? oB0 : q2;  bm[3].y = hi ? oB1 : q3;
        }
      }

      // ---- store ----
      if (pass == 0) {
        // time-major tile layout: ws[g][t*16 + r] (lane halves store 64B runs)
        float* sp = wsg + (size_t)(tb + 8*hi) * 16 + r;
        #pragma unroll
        for (int v = 0; v < 8; ++v) sp[v * 16] = c[v];
      } else {
        // y2 at reversed time t maps to output tau = TEXT-1-PADL - t (reverse + trim)
        const int tbase = tb + 8*hi;
        if (tb >= 48 && tb + 16 <= 30033) {                  // interior: branch-free
          float* op = out + (size_t)row * TLEN + ((TEXT - 1 - PADL) - tbase);
          #pragma unroll
          for (int v = 0; v < 8; ++v) op[-v] = c[v];
        } else {
          #pragma unroll
          for (int v = 0; v < 8; ++v) {
            int tau = (TEXT - 1 - PADL) - (tbase + v);
            if (tau >= 0 && tau < TLEN)
              out[(size_t)row * TLEN + tau] = c[v];
          }
        }
      }
    }
    // forward-pass stores must be visible before same-wave reversed reads
    asm volatile("s_wait_storecnt 0" ::: "memory");
  }
}

// ---------------- host: exact replica of the reference filter design ----------------

static void buildCoefs(Coef& C) {
  const double PI = 3.14159265358979323846;
  const int N = 5;
  const double fs = 500.0, low = 0.5, high = 45.0, fs2 = 2.0;

  std::complex<double> p[5];
  for (int k = 1; k <= N; ++k)
    p[k-1] = std::exp(std::complex<double>(0.0, PI * (2.0*k + N - 1.0) / (2.0*N)));

  double Wn0 = low  / (fs * 0.5), Wn1 = high / (fs * 0.5);
  double w0 = 2.0*fs2*std::tan(PI*Wn0/fs2);
  double w1 = 2.0*fs2*std::tan(PI*Wn1/fs2);
  double bw = w1 - w0, wo = std::sqrt(w0*w1);

  std::complex<double> pbp[10];
  for (int i = 0; i < N; ++i) {
    std::complex<double> ps = p[i] * (bw * 0.5);
    std::complex<double> rt = std::sqrt(ps*ps - std::complex<double>(wo*wo, 0.0));
    pbp[i]     = ps + rt;
    pbp[i + N] = ps - rt;
  }
  double gain = std::pow(bw, (double)N);
  const double f2 = 2.0 * fs2;  // 4.0
  std::complex<double> pd[10], prod(1.0, 0.0);
  for (int i = 0; i < 10; ++i) {
    pd[i] = (f2 + pbp[i]) / (f2 - pbp[i]);
    prod *= (f2 - pbp[i]);
  }
  gain *= std::real(std::pow(f2, (double)N) / prod);

  const double tol = 1e-10;
  std::complex<double> comp[10]; int nc = 0;
  double realp[10]; int nr = 0;
  for (int i = 0; i < 10; ++i) {
    if (std::fabs(pd[i].imag()) < tol) realp[nr++] = pd[i].real();
    else if (pd[i].imag() > tol)       comp[nc++]  = pd[i];
  }
  std::sort(realp, realp + nr);

  std::complex<double> p1s[5], p2s[5]; int s = 0;
  for (int i = 0; i < nc; ++i) { p1s[s] = comp[i]; p2s[s] = std::conj(comp[i]); ++s; }
  for (int i = 0; i < nr/2; ++i) { p1s[s] = realp[2*i]; p2s[s] = realp[2*i+1]; ++s; }

  double sos[5][6];
  for (int i = 0; i < 5; ++i) {
    sos[i][0] = 1.0; sos[i][1] = 0.0; sos[i][2] = -1.0;
    sos[i][3] = 1.0;
    sos[i][4] = -std::real(p1s[i] + p2s[i]);
    sos[i][5] =  std::real(p1s[i] * p2s[i]);
  }
  sos[0][0] *= gain; sos[0][1] *= gain; sos[0][2] *= gain;

  double zi[5][2]; double scale = 1.0;
  for (int k = 0; k < 5; ++k) {
    double bsum = sos[k][0] + sos[k][1] + sos[k][2];
    double asum = sos[k][3] + sos[k][4] + sos[k][5];
    double K  = bsum / asum;
    double zz2 = sos[k][2] - sos[k][5]*K;
    double zz1 = sos[k][1] - sos[k][4]*K + zz2;
    zi[k][0] = scale * zz1; zi[k][1] = scale * zz2;
    scale *= K;
  }

  // block-state-space data (TDF-II: s=(z1,z2), A=[[-a1,1],[-a2,0]],
  // B=[b1-a1*b0, b2-a2*b0], C=[1,0], d=b0)
  for (int k = 0; k < 5; ++k) {
    double b0 = sos[k][0], b1 = sos[k][1], b2 = sos[k][2];
    double a1 = sos[k][4], a2 = sos[k][5];
    double A00 = -a1, A01 = 1.0, A10 = -a2, A11 = 0.0;
    double B0 = b1 - a1*b0, B1 = b2 - a2*b0;
    Sec& S = C.s[k];

    double h[16]; h[0] = b0;
    { double v0 = B0, v1 = B1;
      for (int t = 1; t < 16; ++t) {
        h[t] = v0;
        double n0 = A00*v0 + A01*v1, n1 = A10*v0 + A11*v1;
        v0 = n0; v1 = n1;
      } }
    double g1[16], g2[16];
    double M00 = 1, M01 = 0, M10 = 0, M11 = 1;   // A^0
    for (int t = 0; t < 16; ++t) {
      g1[t] = M00; g2[t] = M01;
      double n00 = A00*M00 + A01*M10, n01 = A00*M01 + A01*M11;
      double n10 = A10*M00 + A11*M10, n11 = A10*M01 + A11*M11;
      M00 = n00; M01 = n01; M10 = n10; M11 = n11;
    }
    for (int t = 0; t < 16; ++t) {
      S.h[t]  = (float)h[t];
      S.g1[t] = (float)g1[t]; S.g2[t] = (float)g2[t];
    }
    S.b1 = (float)b1; S.b2 = (float)b2;
    S.a1 = (float)a1; S.a2 = (float)a2;
    S.zi1 = (float)zi[k][0]; S.zi2 = (float)zi[k][1];
  }
}

extern "C" void kernel_launch(void* const* d_in, const int* in_sizes, int n_in,
                              void* d_out, int out_size, void* d_ws, size_t ws_size,
                              hipStream_t stream) {
  (void)in_sizes; (void)n_in; (void)out_size; (void)ws_size;
  const float* in = (const float*)d_in[0];      // (64,12,30000) f32 -> 768 rows
  float* out = (float*)d_out;                   // same shape f32
  float* ws  = (float*)d_ws;                    // needs NGRP*TPAD*16*4 = ~92.4 MB

  Coef cf;
  buildCoefs(cf);                               // deterministic, tiny (host doubles)

  dim3 grid(NGRP), block(32);                   // one wave32 per 16-row group
  BandPass_filtfilt_wmma<<<grid, block, 0, stream>>>(in, out, ws, cf);
}